// GraphAttentionLayer_5660766896648
// MI455X (gfx1250) — compile-verified
//
#include <hip/hip_runtime.h>
#include <math.h>

// Problem constants (from reference setup_inputs)
constexpr int B     = 8;
constexpr int N     = 2048;
constexpr int IN_F  = 128;
constexpr int OUT_F = 64;

#define ALPHA 0.2f
#define NEGV  -9000000000000000.0f

typedef __attribute__((ext_vector_type(2))) float v2f;
typedef __attribute__((ext_vector_type(8))) float v8f;

// ---------------------------------------------------------------------------
// Kernel 1: Wh = h @ W   (per block: 16 rows x 64 cols; 4 waves, each a 16x16
// tile via V_WMMA_F32_16X16X4_F32, K-loop over 128 in steps of 4)
// ---------------------------------------------------------------------------
__global__ void wh_kernel(const float* __restrict__ h,
                          const float* __restrict__ W,
                          float* __restrict__ Wh) {
  const int lane = threadIdx.x & 31;
  const int wv   = threadIdx.x >> 5;          // 0..3 -> 16-col chunk of OUT_F
  const int tile = blockIdx.x;                // b*(N/16) + mtile
  const int b    = tile / (N / 16);
  const int m0   = (tile % (N / 16)) * 16;
  const int half = lane >> 4;                 // 0: K pair {0,1}, 1: {2,3}
  const int lm   = lane & 15;                 // row (A) / col (B,C)
  const int f0   = wv * 16;

  const float* hrow = h + ((size_t)b * N + (m0 + lm)) * IN_F;

  v8f c = {};
  for (int k = 0; k < IN_F; k += 4) {
    const int ka = k + 2 * half;
    v2f av, bv;
    av.x = hrow[ka];
    av.y = hrow[ka + 1];
    bv.x = W[(size_t)ka * OUT_F + f0 + lm];
    bv.y = W[(size_t)(ka + 1) * OUT_F + f0 + lm];
    c = __builtin_amdgcn_wmma_f32_16x16x4_f32(false, av, false, bv,
                                              (short)0, c, false, false);
  }

  float* outp = Wh + ((size_t)b * N + m0) * OUT_F + f0 + lm;
  for (int r = 0; r < 8; ++r)
    outp[(size_t)(r + 8 * half) * OUT_F] = c[r];
}

// ---------------------------------------------------------------------------
// Kernel 2: s1 = Wh @ a[:64], s2 = Wh @ a[64:]  (thread per row; ~4MB traffic)
// ---------------------------------------------------------------------------
__global__ void s_kernel(const float* __restrict__ Wh,
                         const float* __restrict__ a,
                         float* __restrict__ s1, float* __restrict__ s2) {
  const int i = blockIdx.x * blockDim.x + threadIdx.x; // 0 .. B*N-1
  if (i >= B * N) return;
  const float* row = Wh + (size_t)i * OUT_F;
  float r1 = 0.f, r2 = 0.f;
  for (int f = 0; f < OUT_F; ++f) {
    const float v = row[f];
    r1 += v * a[f];
    r2 += v * a[OUT_F + f];
  }
  s1[i] = r1;
  s2[i] = r2;
}

// ---------------------------------------------------------------------------
// Kernel 3: attention row softmax.
// One block (256 threads) per (b,i) row of 2048 elements; each thread owns 8
// coalesced elements. LeakyReLU + adjacency mask + stable softmax.
// ---------------------------------------------------------------------------
__global__ void attn_kernel(const int* __restrict__ adj,
                            const float* __restrict__ s1,
                            const float* __restrict__ s2,
                            float* __restrict__ attn) {
  const int row = blockIdx.x;            // b*N + i
  const int b   = row / N;
  const int t   = threadIdx.x;           // 0..255
  __shared__ float red[256];

  const int*   adjrow = adj + (size_t)row * N;
  const float* s2b    = s2 + (size_t)b * N;
  const float  my1    = s1[row];

  float e[8];
  float mx = -INFINITY;
  for (int cidx = 0; cidx < 8; ++cidx) {
    const int j = cidx * 256 + t;
    float v = my1 + s2b[j];
    v = (v >= 0.f) ? v : ALPHA * v;
    v = (adjrow[j] > 0) ? v : NEGV;
    e[cidx] = v;
    mx = fmaxf(mx, v);
  }

  red[t] = mx;
  __syncthreads();
  for (int s = 128; s > 0; s >>= 1) {
    if (t < s) red[t] = fmaxf(red[t], red[t + s]);
    __syncthreads();
  }
  mx = red[0];
  __syncthreads();

  float sum = 0.f;
  for (int cidx = 0; cidx < 8; ++cidx) {
    e[cidx] = expf(e[cidx] - mx);
    sum += e[cidx];
  }
  red[t] = sum;
  __syncthreads();
  for (int s = 128; s > 0; s >>= 1) {
    if (t < s) red[t] += red[t + s];
    __syncthreads();
  }
  const float inv = 1.0f / red[0];

  float* arow = attn + (size_t)row * N;
  for (int cidx = 0; cidx < 8; ++cidx)
    arow[cidx * 256 + t] = e[cidx] * inv;
}

// ---------------------------------------------------------------------------
// Kernel 4: out = ELU'(attn @ Wh).  One wave per 16-row tile; 4 f32 C tiles
// cover the full 64-wide N; A tile loaded once per K-step, reused 4x.
// K-loop: 2048/4 = 512 WMMA steps x 4 tiles.
// ---------------------------------------------------------------------------
__global__ void out_kernel(const float* __restrict__ attn,
                           const float* __restrict__ Wh,
                           float* __restrict__ out) {
  const int lane = threadIdx.x & 31;
  const int tile = blockIdx.x;               // b*(N/16) + mtile
  const int b    = tile / (N / 16);
  const int m0   = (tile % (N / 16)) * 16;
  const int half = lane >> 4;
  const int lm   = lane & 15;

  const float* arow = attn + ((size_t)b * N + m0 + lm) * N;
  const float* Whb  = Wh + (size_t)b * N * OUT_F;

  v8f c0 = {}, c1 = {}, c2 = {}, c3 = {};
  for (int k = 0; k < N; k += 4) {
    const int ka = k + 2 * half;
    v2f av;
    av.x = arow[ka];
    av.y = arow[ka + 1];
    const float* w0 = Whb + (size_t)ka * OUT_F + lm;
    const float* w1 = w0 + OUT_F;
    v2f b0; b0.x = w0[0];  b0.y = w1[0];
    v2f b1; b1.x = w0[16]; b1.y = w1[16];
    v2f b2; b2.x = w0[32]; b2.y = w1[32];
    v2f b3; b3.x = w0[48]; b3.y = w1[48];
    c0 = __builtin_amdgcn_wmma_f32_16x16x4_f32(false, av, false, b0, (short)0, c0, false, false);
    c1 = __builtin_amdgcn_wmma_f32_16x16x4_f32(false, av, false, b1, (short)0, c1, false, false);
    c2 = __builtin_amdgcn_wmma_f32_16x16x4_f32(false, av, false, b2, (short)0, c2, false, false);
    c3 = __builtin_amdgcn_wmma_f32_16x16x4_f32(false, av, false, b3, (short)0, c3, false, false);
  }

  float* orow = out + ((size_t)b * N + m0) * OUT_F;
  for (int r = 0; r < 8; ++r) {
    const int m = r + 8 * half;
    float v;
    v = c0[r]; orow[(size_t)m * OUT_F +  0 + lm] = (v > 0.f) ? v : expm1f(v);
    v = c1[r]; orow[(size_t)m * OUT_F + 16 + lm] = (v > 0.f) ? v : expm1f(v);
    v = c2[r]; orow[(size_t)m * OUT_F + 32 + lm] = (v > 0.f) ? v : expm1f(v);
    v = c3[r]; orow[(size_t)m * OUT_F + 48 + lm] = (v > 0.f) ? v : expm1f(v);
  }
}

// ---------------------------------------------------------------------------
extern "C" void kernel_launch(void* const* d_in, const int* in_sizes, int n_in,
                              void* d_out, int out_size, void* d_ws, size_t ws_size,
                              hipStream_t stream) {
  const float* h   = (const float*)d_in[0];   // (B, N, IN_F)
  const int*   adj = (const int*)  d_in[1];   // (B, N, N)
  const float* W   = (const float*)d_in[2];   // (IN_F, OUT_F)
  const float* a   = (const float*)d_in[3];   // (2*OUT_F, 1)

  float* out_p  = (float*)d_out;                        // (B, N, OUT_F)
  float* attn_p = out_p + (size_t)B * N * OUT_F;        // (B, N, N)

  float* Wh = (float*)d_ws;                             // B*N*OUT_F floats
  float* s1 = Wh + (size_t)B * N * OUT_F;               // B*N floats
  float* s2 = s1 + (size_t)B * N;                       // B*N floats

  wh_kernel  <<<B * (N / 16), 128, 0, stream>>>(h, W, Wh);
  s_kernel   <<<(B * N) / 256, 256, 0, stream>>>(Wh, a, s1, s2);
  attn_kernel<<<B * N,         256, 0, stream>>>(adj, s1, s2, attn_p);
  out_kernel <<<B * (N / 16),   32, 0, stream>>>(attn_p, Wh, out_p);
}